// SimpleBlock2d_5153960755514
// MI455X (gfx1250) — compile-verified
//
#include <hip/hip_runtime.h>
#include <math.h>

// ---------------------------------------------------------------------------
// FNO-3D forward for MI455X (gfx1250, wave32). The z-DFT stages, pointwise conv
// and fc1 run on V_WMMA_F32_16X16X4_F32; x/y DFTs and the spectral mix are scalar.
// ---------------------------------------------------------------------------

typedef float v2f __attribute__((ext_vector_type(2)));
typedef float v8f __attribute__((ext_vector_type(8)));

__device__ __forceinline__ v8f wmma4(v2f a, v2f b, v8f c) {
  // (neg_a, A, neg_b, B, c_mod, C, reuse_a, reuse_b)
  return __builtin_amdgcn_wmma_f32_16x16x4_f32(false, a, false, b, (short)0, c,
                                               false, false);
}

#define CH 32
#define VOL 262144            // 64^3
#define HTOT 33554432         // 4*32*64^3

// workspace offsets (floats)
static const size_t OFF_H     = 0;
static const size_t OFF_T     = 33554432;
static const size_t OFF_ZF    = 67108864;   // [B][C][64x][16m][64y]
static const size_t OFF_YF    = 75497472;   // [B][C][16jy][8kz][2][64x]
static const size_t OFF_XFB   = 77594624;   // [B][C][16jx][16jy][8kz][2]
static const size_t OFF_XFB2  = 78118912;
static const size_t OFF_YF2   = 78643200;
static const size_t OFF_ZF2   = 80740352;   // [B][C][64x][16k][64y]
static const size_t OFF_PART  = 89128960;   // [32][64][2]
static const size_t OFF_STATS = 89133056;   // [32][2] (mean, inv_std)
static const size_t OFF_TZF   = 89133120;   // [64z][16m] forward z-DFT
static const size_t OFF_TZI   = 89134144;   // [16k][64z] inverse z-DFT (real)
static const size_t OFF_TABF  = 89135168;   // [64n][16j][2] forward x/y exp
static const size_t OFF_TABI  = 89137216;   // [16j][64n][2] inverse x/y exp

// ---------------------------------------------------------------------------
__global__ void fno_tables(float* __restrict__ tzf, float* __restrict__ tzi,
                           float* __restrict__ tabf, float* __restrict__ tabi) {
  const int tid = threadIdx.x;
  const double w = 6.283185307179586476925286766559 / 64.0;
  for (int idx = tid; idx < 1024; idx += 256) {      // tzf[z][m16]
    int n = idx >> 4, j = idx & 15;
    tzf[idx] = (j < 8) ? (float)(cos(w * j * n) * 0.125)
                       : (float)(-sin(w * (j - 8) * n) * 0.125);
  }
  for (int idx = tid; idx < 1024; idx += 256) {      // tzi[k16][z]
    int k16 = idx >> 6, n = idx & 63, k = k16 & 7;
    double ck = (k == 0) ? 1.0 : 2.0;
    tzi[idx] = (k16 < 8) ? (float)(ck * cos(w * k * n) * 0.125)
                         : (float)(-ck * sin(w * k * n) * 0.125);
  }
  for (int idx = tid; idx < 2048; idx += 256) {      // tabf[n][j][c,s]
    int n = idx >> 5, j = (idx >> 1) & 15, ri = idx & 1;
    int kj = (j < 8) ? j : j - 16;
    double th = w * kj * n;
    tabf[idx] = ri ? (float)(sin(th) * 0.125) : (float)(cos(th) * 0.125);
  }
  for (int idx = tid; idx < 2048; idx += 256) {      // tabi[j][n][c,s]
    int j = idx >> 7, n = (idx >> 1) & 63, ri = idx & 1;
    int kj = (j < 8) ? j : j - 16;
    double th = w * kj * n;
    tabi[idx] = ri ? (float)(sin(th) * 0.125) : (float)(cos(th) * 0.125);
  }
}

// ---------------------------------------------------------------------------
__global__ void fno_lift(const float* __restrict__ xin,
                         const float* __restrict__ w,
                         const float* __restrict__ bias,
                         float* __restrict__ h) {
  int idx = blockIdx.x * 256 + threadIdx.x;          // [b][c][x][y][z]
  int z = idx & 63, y = (idx >> 6) & 63, x = (idx >> 12) & 63;
  int c = (idx >> 18) & 31, b = idx >> 23;
  size_t vox = ((size_t)(b * 64 + x) * 64 + y) * 64 + z;
  const float* xp = xin + vox * 3;
  h[idx] = xp[0] * w[c * 3] + xp[1] * w[c * 3 + 1] + xp[2] * w[c * 3 + 2] +
           bias[c];
}

// ---------------------------------------------------------------------------
// Forward z-DFT: D[m=mode16][n=y16tile] = sum_z Tzf[z][m] * h[b,c,x,y,z]
__global__ void fno_zdft(const float* __restrict__ h,
                         const float* __restrict__ tzf,
                         float* __restrict__ zf) {
  __shared__ float s_t[1024];
  const int tid = threadIdx.x;
  int x = blockIdx.x & 63, c = (blockIdx.x >> 6) & 31, b = blockIdx.x >> 11;
  for (int i = tid; i < 1024; i += 128) s_t[i] = tzf[i];
  __syncthreads();

  const int lane = tid & 31, row = lane & 15, half = lane >> 4;
  const int y0 = (tid >> 5) * 16;
  const float* hb =
      h + ((size_t)(b * 32 + c)) * VOL + (size_t)x * 4096 + (size_t)y0 * 64;
  v8f acc = {0.f, 0.f, 0.f, 0.f, 0.f, 0.f, 0.f, 0.f};
#pragma unroll
  for (int kk = 0; kk < 16; ++kk) {
    int kb = kk * 4 + 2 * half;
    v2f a, bb;
    a.x = s_t[kb * 16 + row];
    a.y = s_t[(kb + 1) * 16 + row];
    const float* hp = hb + (size_t)row * 64 + kb;
    bb.x = hp[0];
    bb.y = hp[1];
    acc = wmma4(a, bb, acc);
  }
  float* zb = zf + ((size_t)(b * 32 + c) * 64 + x) * 1024;
#pragma unroll
  for (int v = 0; v < 8; ++v) zb[(v + 8 * half) * 64 + y0 + row] = acc[v];
}

// ---------------------------------------------------------------------------
// Forward y-DFT (complex, 16 modes), per (b,c,x) line.
__global__ void fno_ydft(const float* __restrict__ zf,
                         const float* __restrict__ tabf,
                         float* __restrict__ yf) {
  __shared__ float s_z[1024];
  __shared__ float s_tf[2048];
  const int tid = threadIdx.x;
  int x = blockIdx.x & 63, c = (blockIdx.x >> 6) & 31, b = blockIdx.x >> 11;
  const float* zb = zf + ((size_t)(b * 32 + c) * 64 + x) * 1024;
  for (int i = tid; i < 1024; i += 128) s_z[i] = zb[i];
  for (int i = tid; i < 2048; i += 128) s_tf[i] = tabf[i];
  __syncthreads();

  int j = tid >> 3, kz = tid & 7;
  float sr = 0.f, si = 0.f;
#pragma unroll 4
  for (int y = 0; y < 64; ++y) {
    float ar = s_z[kz * 64 + y], ai = s_z[(kz + 8) * 64 + y];
    float cc = s_tf[y * 32 + j * 2], ss = s_tf[y * 32 + j * 2 + 1];
    sr += ar * cc + ai * ss;   // e^{-i th}
    si += ai * cc - ar * ss;
  }
  size_t base = (size_t)(b * 32 + c) * 16384 + (size_t)j * 1024 + kz * 128;
  yf[base + x] = sr;
  yf[base + 64 + x] = si;
}

// ---------------------------------------------------------------------------
// Forward x-DFT, per (b,c,kz).
__global__ void fno_xdft(const float* __restrict__ yf,
                         const float* __restrict__ tabf,
                         float* __restrict__ xfb) {
  __shared__ float s_y[2048];
  __shared__ float s_tf[2048];
  const int tid = threadIdx.x;
  int kz = blockIdx.x & 7, c = (blockIdx.x >> 3) & 31, b = blockIdx.x >> 8;
  size_t base = (size_t)(b * 32 + c) * 16384 + kz * 128;
  for (int i = tid; i < 2048; i += 256) {
    int jy = i >> 7, r = i & 127;
    s_y[i] = yf[base + (size_t)jy * 1024 + r];
  }
  for (int i = tid; i < 2048; i += 256) s_tf[i] = tabf[i];
  __syncthreads();

  int jx = tid >> 4, jy = tid & 15;
  float sr = 0.f, si = 0.f;
#pragma unroll 4
  for (int x = 0; x < 64; ++x) {
    float ar = s_y[jy * 128 + x], ai = s_y[jy * 128 + 64 + x];
    float cc = s_tf[x * 32 + jx * 2], ss = s_tf[x * 32 + jx * 2 + 1];
    sr += ar * cc + ai * ss;
    si += ai * cc - ar * ss;
  }
  size_t o = (size_t)(b * 32 + c) * 4096 + (jx * 16 + jy) * 16 + kz * 2;
  xfb[o] = sr;
  xfb[o + 1] = si;
}

// ---------------------------------------------------------------------------
// Spectral channel mix: out[o] = sum_i in[i] * wc[corner][i][o] (complex)
__global__ void fno_spectral(const float* __restrict__ xfb,
                             const float* __restrict__ sw, int layer,
                             float* __restrict__ xfb2) {
  __shared__ float s_w[8192];   // [i32][ol16][mzri16]
  __shared__ float s_x[512];    // [i32][mzri16]
  const int tid = threadIdx.x;
  int oh = blockIdx.x & 1, jy = (blockIdx.x >> 1) & 15,
      jx = (blockIdx.x >> 5) & 15, b = blockIdx.x >> 9;
  int corner = (jx >= 8 ? 1 : 0) + (jy >= 8 ? 2 : 0);
  int mx = jx & 7, my = jy & 7;
  size_t wbase = (size_t)(layer * 4 + corner) * 32;  // i index follows
  for (int it = 0; it < 64; ++it) {
    int f = tid + 128 * it;
    int i = f >> 8, ol = (f >> 4) & 15, mzri = f & 15;
    int o = oh * 16 + ol;
    s_w[f] = sw[((wbase + i) * 32 + o) * 1024 + (mx * 8 + my) * 16 + mzri];
  }
  for (int it = 0; it < 4; ++it) {
    int f = tid + 128 * it;
    int i = f >> 4, mzri = f & 15;
    s_x[f] = xfb[(size_t)(b * 32 + i) * 4096 + (jx * 16 + jy) * 16 + mzri];
  }
  __syncthreads();

  int ol = tid >> 3, kz = tid & 7, o = oh * 16 + ol;
  float sr = 0.f, si = 0.f;
#pragma unroll
  for (int i = 0; i < 32; ++i) {
    float ar = s_x[i * 16 + kz * 2], ai = s_x[i * 16 + kz * 2 + 1];
    float wr = s_w[(i * 16 + ol) * 16 + kz * 2];
    float wi = s_w[(i * 16 + ol) * 16 + kz * 2 + 1];
    sr += ar * wr - ai * wi;
    si += ar * wi + ai * wr;
  }
  size_t oidx = (size_t)(b * 32 + o) * 4096 + (jx * 16 + jy) * 16 + kz * 2;
  xfb2[oidx] = sr;
  xfb2[oidx + 1] = si;
}

// ---------------------------------------------------------------------------
// Inverse x-DFT, per (b,o).
__global__ void fno_xinv(const float* __restrict__ xfb2,
                         const float* __restrict__ tabi,
                         float* __restrict__ yf2) {
  __shared__ float s_x2[4096];
  __shared__ float s_ti[2048];
  const int tid = threadIdx.x;
  int o = blockIdx.x & 31, b = blockIdx.x >> 5;
  const float* xb = xfb2 + (size_t)(b * 32 + o) * 4096;
  for (int i = tid; i < 4096; i += 256) s_x2[i] = xb[i];
  for (int i = tid; i < 2048; i += 256) s_ti[i] = tabi[i];
  __syncthreads();

  int pair = tid >> 1, jy = pair >> 3, kz = pair & 7, x0 = (tid & 1) * 32;
  size_t base = (size_t)(b * 32 + o) * 16384 + (size_t)jy * 1024 + kz * 128;
  for (int x = x0; x < x0 + 32; ++x) {
    float sr = 0.f, si = 0.f;
#pragma unroll
    for (int jx = 0; jx < 16; ++jx) {
      float ar = s_x2[(jx * 16 + jy) * 16 + kz * 2];
      float ai = s_x2[(jx * 16 + jy) * 16 + kz * 2 + 1];
      float cc = s_ti[jx * 128 + x * 2], ss = s_ti[jx * 128 + x * 2 + 1];
      sr += ar * cc - ai * ss;   // e^{+i th}
      si += ar * ss + ai * cc;
    }
    yf2[base + x] = sr;
    yf2[base + 64 + x] = si;
  }
}

// ---------------------------------------------------------------------------
// Inverse y-DFT, per (b,o,x); packs 16 z-mode (r,i) values per y.
__global__ void fno_yinv(const float* __restrict__ yf2,
                         const float* __restrict__ tabi,
                         float* __restrict__ zf2) {
  __shared__ float s_y2[256];   // [jy16][kzri16]
  __shared__ float s_ti[2048];
  const int tid = threadIdx.x;
  int x = blockIdx.x & 63, o = (blockIdx.x >> 6) & 31, b = blockIdx.x >> 11;
  size_t base = (size_t)(b * 32 + o) * 16384;
  for (int i = tid; i < 256; i += 128) {
    int jy = i >> 4, kzri = i & 15;
    s_y2[i] = yf2[base + (size_t)jy * 1024 + kzri * 64 + x];
  }
  for (int i = tid; i < 2048; i += 128) s_ti[i] = tabi[i];
  __syncthreads();

  int k16 = tid >> 3, y0 = (tid & 7) * 8;
  int kz = k16 & 7, im = (k16 >= 8);
  float* zb = zf2 + ((size_t)(b * 32 + o) * 64 + x) * 1024 + k16 * 64;
  for (int y = y0; y < y0 + 8; ++y) {
    float s = 0.f;
#pragma unroll
    for (int jy = 0; jy < 16; ++jy) {
      float ar = s_y2[jy * 16 + kz * 2], ai = s_y2[jy * 16 + kz * 2 + 1];
      float cc = s_ti[jy * 128 + y * 2], ss = s_ti[jy * 128 + y * 2 + 1];
      s += im ? (ar * ss + ai * cc) : (ar * cc - ai * ss);
    }
    zb[y] = s;
  }
}

// ---------------------------------------------------------------------------
// Fused: t = inverse-z-DFT(zf2) + pointwise(h, cw) + cb.  WMMA: M=o, N=z.
__global__ void fno_layer_mix(const float* __restrict__ h,
                              const float* __restrict__ zf2,
                              const float* __restrict__ tzi,
                              const float* __restrict__ cw,
                              const float* __restrict__ cb, int layer,
                              float* __restrict__ t) {
  __shared__ float s_cw[1024];  // [o][c]
  __shared__ float s_tz[1024];  // [k16][z]
  __shared__ float s_zs[512];   // [o][k16]
  __shared__ float s_cb[32];
  const int tid = threadIdx.x;
  int y = blockIdx.x & 63, x = (blockIdx.x >> 6) & 63, b = blockIdx.x >> 12;
  for (int i = tid; i < 1024; i += 128) s_cw[i] = cw[layer * 1024 + i];
  for (int i = tid; i < 1024; i += 128) s_tz[i] = tzi[i];
  for (int i = tid; i < 512; i += 128) {
    int o = i >> 4, k = i & 15;
    s_zs[i] = zf2[((size_t)(b * 32 + o) * 64 + x) * 1024 + k * 64 + y];
  }
  if (tid < 32) s_cb[tid] = cb[layer * 32 + tid];
  __syncthreads();

  const int lane = tid & 31, row = lane & 15, half = lane >> 4;
  const int z0 = (tid >> 5) * 16;
  const float* hb = h + (size_t)b * 32 * VOL + (size_t)x * 4096 +
                    (size_t)y * 64 + z0 + row;
#pragma unroll
  for (int mt = 0; mt < 2; ++mt) {
    int o0 = mt * 16;
    v8f acc;
#pragma unroll
    for (int v = 0; v < 8; ++v) acc[v] = s_cb[o0 + v + 8 * half];
    // pointwise conv: K = 32 channels
#pragma unroll
    for (int kk = 0; kk < 8; ++kk) {
      int kb = kk * 4 + 2 * half;
      v2f a, bb;
      a.x = s_cw[(o0 + row) * 32 + kb];
      a.y = s_cw[(o0 + row) * 32 + kb + 1];
      bb.x = hb[(size_t)kb * VOL];
      bb.y = hb[(size_t)(kb + 1) * VOL];
      acc = wmma4(a, bb, acc);
    }
    // inverse z-DFT: K = 16 packed (r,i) modes
#pragma unroll
    for (int kk = 0; kk < 4; ++kk) {
      int kb = kk * 4 + 2 * half;
      v2f a, bb;
      a.x = s_zs[(o0 + row) * 16 + kb];
      a.y = s_zs[(o0 + row) * 16 + kb + 1];
      bb.x = s_tz[kb * 64 + z0 + row];
      bb.y = s_tz[(kb + 1) * 64 + z0 + row];
      acc = wmma4(a, bb, acc);
    }
#pragma unroll
    for (int v = 0; v < 8; ++v) {
      int o = o0 + v + 8 * half;
      t[(size_t)(b * 32 + o) * VOL + (size_t)x * 4096 + (size_t)y * 64 + z0 +
        row] = acc[v];
    }
  }
}

// ---------------------------------------------------------------------------
__global__ void fno_bn_part(const float* __restrict__ t,
                            float* __restrict__ part) {
  __shared__ float r1[256];
  __shared__ float r2[256];
  const int tid = threadIdx.x;
  int c = blockIdx.x >> 6, p = blockIdx.x & 63;
  float s1 = 0.f, s2 = 0.f;
  for (int b = 0; b < 4; ++b) {
    const float* tb = t + (size_t)(b * 32 + c) * VOL + (size_t)p * 4096;
    for (int j = tid; j < 4096; j += 256) {
      float v = tb[j];
      s1 += v;
      s2 += v * v;
    }
  }
  r1[tid] = s1;
  r2[tid] = s2;
  __syncthreads();
  for (int off = 128; off > 0; off >>= 1) {
    if (tid < off) {
      r1[tid] += r1[tid + off];
      r2[tid] += r2[tid + off];
    }
    __syncthreads();
  }
  if (tid == 0) {
    part[(c * 64 + p) * 2] = r1[0];
    part[(c * 64 + p) * 2 + 1] = r2[0];
  }
}

__global__ void fno_bn_final(const float* __restrict__ part,
                             float* __restrict__ stats) {
  int c = threadIdx.x;
  float s1 = 0.f, s2 = 0.f;
  for (int p = 0; p < 64; ++p) {
    s1 += part[(c * 64 + p) * 2];
    s2 += part[(c * 64 + p) * 2 + 1];
  }
  const float invN = 1.f / 1048576.f;
  float m = s1 * invN;
  float var = s2 * invN - m * m;
  stats[c * 2] = m;
  stats[c * 2 + 1] = rsqrtf(var + 1e-5f);
}

__global__ void fno_bn_apply(const float* __restrict__ t,
                             const float* __restrict__ stats,
                             const float* __restrict__ gamma,
                             const float* __restrict__ beta, int layer,
                             int do_relu, float* __restrict__ h) {
  int idx = blockIdx.x * 256 + threadIdx.x;
  int c = (idx >> 18) & 31;
  float v = (t[idx] - stats[c * 2]) * stats[c * 2 + 1] * gamma[layer * 32 + c] +
            beta[layer * 32 + c];
  if (do_relu) v = fmaxf(v, 0.f);
  h[idx] = v;
}

// ---------------------------------------------------------------------------
// Fused head: out = fc2( relu( fc1(h) + b1 ) ) + b2.  WMMA: M=f-tile, N=z.
__global__ void fno_fc_head(const float* __restrict__ h,
                            const float* __restrict__ w1,
                            const float* __restrict__ b1,
                            const float* __restrict__ w2,
                            const float* __restrict__ b2,
                            float* __restrict__ out) {
  __shared__ float s_w1[4096];  // [f128][c32]
  __shared__ float s_b1[128];
  __shared__ float s_w2[128];
  const int tid = threadIdx.x;
  int y = blockIdx.x & 63, x = (blockIdx.x >> 6) & 63, b = blockIdx.x >> 12;
  for (int i = tid; i < 4096; i += 128) s_w1[i] = w1[i];
  if (tid < 128) {
    s_b1[tid] = b1[tid];
    s_w2[tid] = w2[tid];
  }
  __syncthreads();

  const int lane = tid & 31, row = lane & 15, half = lane >> 4;
  const int z0 = (tid >> 5) * 16;
  const float* hb = h + (size_t)b * 32 * VOL + (size_t)x * 4096 +
                    (size_t)y * 64 + z0 + row;
  float partial = 0.f;
#pragma unroll
  for (int g = 0; g < 8; ++g) {
    v8f acc;
#pragma unroll
    for (int v = 0; v < 8; ++v) acc[v] = s_b1[g * 16 + v + 8 * half];
#pragma unroll
    for (int kk = 0; kk < 8; ++kk) {
      int kb = kk * 4 + 2 * half;
      v2f a, bb;
      a.x = s_w1[(g * 16 + row) * 32 + kb];
      a.y = s_w1[(g * 16 + row) * 32 + kb + 1];
      bb.x = hb[(size_t)kb * VOL];
      bb.y = hb[(size_t)(kb + 1) * VOL];
      acc = wmma4(a, bb, acc);
    }
#pragma unroll
    for (int v = 0; v < 8; ++v) {
      float val = fmaxf(acc[v], 0.f);
      partial += val * s_w2[g * 16 + v + 8 * half];
    }
  }
  float other = __shfl_xor(partial, 16, 32);
  if (half == 0) {
    out[((size_t)(b * 64 + x) * 64 + y) * 64 + z0 + row] =
        partial + other + b2[0];
  }
}

// ---------------------------------------------------------------------------
extern "C" void kernel_launch(void* const* d_in, const int* in_sizes, int n_in,
                              void* d_out, int out_size, void* d_ws,
                              size_t ws_size, hipStream_t stream) {
  (void)in_sizes; (void)n_in; (void)out_size; (void)ws_size;
  const float* xin   = (const float*)d_in[0];
  const float* fc0_w = (const float*)d_in[1];
  const float* fc0_b = (const float*)d_in[2];
  const float* sw    = (const float*)d_in[3];
  const float* cw    = (const float*)d_in[4];
  const float* cb    = (const float*)d_in[5];
  const float* gamma = (const float*)d_in[6];
  const float* beta  = (const float*)d_in[7];
  const float* fc1_w = (const float*)d_in[8];
  const float* fc1_b = (const float*)d_in[9];
  const float* fc2_w = (const float*)d_in[10];
  const float* fc2_b = (const float*)d_in[11];
  float* out = (float*)d_out;
  float* ws = (float*)d_ws;

  float* H    = ws + OFF_H;
  float* T    = ws + OFF_T;
  float* ZF   = ws + OFF_ZF;
  float* YF   = ws + OFF_YF;
  float* XFB  = ws + OFF_XFB;
  float* XFB2 = ws + OFF_XFB2;
  float* YF2  = ws + OFF_YF2;
  float* ZF2  = ws + OFF_ZF2;
  float* PART = ws + OFF_PART;
  float* STAT = ws + OFF_STATS;
  float* TZF  = ws + OFF_TZF;
  float* TZI  = ws + OFF_TZI;
  float* TABF = ws + OFF_TABF;
  float* TABI = ws + OFF_TABI;

  fno_tables<<<1, 256, 0, stream>>>(TZF, TZI, TABF, TABI);
  fno_lift<<<HTOT / 256, 256, 0, stream>>>(xin, fc0_w, fc0_b, H);

  for (int l = 0; l < 4; ++l) {
    fno_zdft<<<8192, 128, 0, stream>>>(H, TZF, ZF);
    fno_ydft<<<8192, 128, 0, stream>>>(ZF, TABF, YF);
    fno_xdft<<<1024, 256, 0, stream>>>(YF, TABF, XFB);
    fno_spectral<<<2048, 128, 0, stream>>>(XFB, sw, l, XFB2);
    fno_xinv<<<128, 256, 0, stream>>>(XFB2, TABI, YF2);
    fno_yinv<<<8192, 128, 0, stream>>>(YF2, TABI, ZF2);
    fno_layer_mix<<<16384, 128, 0, stream>>>(H, ZF2, TZI, cw, cb, l, T);
    fno_bn_part<<<2048, 256, 0, stream>>>(T, PART);
    fno_bn_final<<<1, 32, 0, stream>>>(PART, STAT);
    fno_bn_apply<<<HTOT / 256, 256, 0, stream>>>(T, STAT, gamma, beta, l,
                                                 (l < 3) ? 1 : 0, H);
  }
  fno_fc_head<<<16384, 128, 0, stream>>>(H, fc1_w, fc1_b, fc2_w, fc2_b, out);
}